// CLUBLoss_8976481649059
// MI455X (gfx1250) — compile-verified
//
#include <hip/hip_runtime.h>
#include <cstdint>
#include <cstddef>

// Problem constants (from reference): B=16, D=128, H=64, W=64
#define D_        128
#define HW_       4096      // H*W
#define N_        65536     // B*H*W rows
#define TILE_ROWS 64
#define NTILES    (N_ / TILE_ROWS)   // 1024
#define K1_WGS    512
#define TILES_PER_WG (NTILES / K1_WGS) // 2
#define REC_STRIDE 520               // floats per partial record (513 used)
#define NREC2     64

typedef float v4f __attribute__((ext_vector_type(4)));

// LDS byte-offset of a __shared__ object (generic -> addrspace(3) -> int).
__device__ __forceinline__ unsigned lds_offset_of(void* p) {
    return (unsigned)(unsigned long long)(__attribute__((address_space(3))) void*)p;
}

// ---------------- Kernel 1: one-pass partials ----------------
// Each WG: 2 tiles of 64 rows x 128 channels, double-buffered through LDS.
//   Stage:   async DMA x tile (global, channel-major) -> LDS, XOR-swizzled 16B chunks.
//   Compute: non-temporal b128 reads of mu/logvar, x from LDS (transposed view),
//            accumulate S1,S2,A,Bm per channel in registers + scalar C.
// Record layout in ws: [0..127]=S1, [128..255]=S2, [256..383]=A, [384..511]=Bm, [512]=C.
__global__ __launch_bounds__(256) void club_k1(const float* __restrict__ x,
                                               const float* __restrict__ mu,
                                               const float* __restrict__ lv,
                                               float* __restrict__ ws) {
    // 64KB arena: two 32KB tile buffers; reduction scratch overlays buffer 0
    // after the final compute barrier (never live simultaneously).
    __shared__ __align__(16) char arena[65536];

    const int tid = threadIdx.x;
    const int wv  = tid >> 5;           // wave id 0..7
    const int ln  = tid & 31;           // lane 0..31
    const int d0  = ln << 2;            // this thread's 4 channels d0..d0+3

    float s1[4] = {0.f,0.f,0.f,0.f};
    float s2[4] = {0.f,0.f,0.f,0.f};
    float sa[4] = {0.f,0.f,0.f,0.f};
    float sb[4] = {0.f,0.f,0.f,0.f};
    float cacc  = 0.f;

    const unsigned ldsArena = lds_offset_of(&arena[0]);
    const unsigned long long xbase = (unsigned long long)(uintptr_t)x;

    // Issue the 8 per-thread async b128 chunk loads for one 64x128 x-tile.
    auto issueTile = [&](int buf, int tile) {
        const int b   = tile >> 6;           // 64 tiles per batch image
        const int hw0 = (tile & 63) << 6;    // hw start
        const unsigned base = ldsArena + (unsigned)(buf << 15);
        #pragma unroll
        for (int jj = 0; jj < 8; ++jj) {
            const int g  = tid + (jj << 8);
            const int d  = g >> 4;           // channel 0..127
            const int cw = g & 15;           // 16B chunk within 64-float row
            // global byte offset of x[b][d][hw0 + 4*cw]
            const int goff = ((((b << 7) + d) << 12) + hw0 + (cw << 2)) << 2;
            // swizzled LDS byte offset: row d (256B) + permuted chunk
            const unsigned loff =
                base + (unsigned)((d << 8) + ((cw ^ ((d >> 2) & 15)) << 4));
            asm volatile("global_load_async_to_lds_b128 %0, %1, %2"
                         :: "v"(loff), "v"(goff), "s"(xbase)
                         : "memory");
        }
    };

    const int tile0 = blockIdx.x * TILES_PER_WG;
    issueTile(0, tile0);                     // prologue: stage tile 0

    #pragma unroll
    for (int tt = 0; tt < TILES_PER_WG; ++tt) {
        if (tt + 1 < TILES_PER_WG) {
            issueTile((tt + 1) & 1, tile0 + tt + 1);   // overlap next DMA
            // async loads complete in order: <=8 outstanding => current tile done
            asm volatile("s_wait_asynccnt 0x8" ::: "memory");
        } else {
            asm volatile("s_wait_asynccnt 0x0" ::: "memory");
        }
        __syncthreads();

        const float* tileF = (const float*)(arena + ((tt & 1) << 15));
        const int row0 = (tile0 + tt) << 6;

        #pragma unroll
        for (int j = 0; j < 8; ++j) {
            const int r   = wv + (j << 3);       // local row 0..63
            const int row = row0 + r;            // global row
            const v4f m4 = __builtin_nontemporal_load(
                (const v4f*)(mu + (size_t)row * D_ + d0));
            const v4f l4 = __builtin_nontemporal_load(
                (const v4f*)(lv + (size_t)row * D_ + d0));
            #pragma unroll
            for (int k = 0; k < 4; ++k) {
                const int d = d0 + k;
                // transposed read through the XOR swizzle (<=2-way bank conflict)
                const int fidx = (d << 6) + (((r >> 2) ^ ((d >> 2) & 15)) << 2) + (r & 3);
                const float xv = tileF[fidx];
                const float mk = m4[k];
                const float iv = __expf(-l4[k]);
                cacc  += iv * xv * (xv - 2.0f * mk);
                s1[k] += xv;
                s2[k] += xv * xv;
                sa[k] += iv;
                sb[k] += iv * mk;
            }
        }
        __syncthreads();  // all reads of this buffer done before it is refilled
    }

    // ---- Cross-wave combine (deterministic fixed-order sums) ----
    // Safe to overlay buffer 0: last compute barrier has retired all tile reads.
    float* sP = (float*)arena;                 // [4][128][8] = 16 KB
    float* sC = (float*)(arena + 16384);       // [256]       =  1 KB
    #pragma unroll
    for (int k = 0; k < 4; ++k) {
        const int d = d0 + k;
        sP[(0 * 128 + d) * 8 + wv] = s1[k];
        sP[(1 * 128 + d) * 8 + wv] = s2[k];
        sP[(2 * 128 + d) * 8 + wv] = sa[k];
        sP[(3 * 128 + d) * 8 + wv] = sb[k];
    }
    sC[tid] = cacc;
    __syncthreads();

    float* rec = ws + (size_t)blockIdx.x * REC_STRIDE;
    if (tid < 128) {
        #pragma unroll
        for (int arr = 0; arr < 4; ++arr) {
            float s = 0.f;
            #pragma unroll
            for (int w = 0; w < 8; ++w) s += sP[(arr * 128 + tid) * 8 + w];
            rec[arr * 128 + tid] = s;
        }
    }
    for (int s = 128; s > 0; s >>= 1) {
        if (tid < s) sC[tid] += sC[tid + s];
        __syncthreads();
    }
    if (tid == 0) rec[512] = sC[0];
}

// ---------------- Kernel 2: fold 8 records -> 1 (x64 WGs) ----------------
__global__ __launch_bounds__(256) void club_k2(const float* __restrict__ in,
                                               float* __restrict__ out) {
    const int g = blockIdx.x;                     // 0..63
    const float* base = in + (size_t)g * 8 * REC_STRIDE;
    for (int idx = threadIdx.x; idx < 513; idx += 256) {
        float s = 0.f;
        #pragma unroll
        for (int r = 0; r < 8; ++r) s += base[r * REC_STRIDE + idx];
        out[(size_t)g * REC_STRIDE + idx] = s;
    }
}

// ---------------- Kernel 3: final 64 records -> scalar ----------------
__global__ __launch_bounds__(512) void club_k3(const float* __restrict__ in,
                                               float* __restrict__ out) {
    __shared__ float S[512];
    __shared__ float T[128];
    __shared__ float Cs;
    const int tid = threadIdx.x;

    float s = 0.f;
    for (int r = 0; r < NREC2; ++r) s += in[(size_t)r * REC_STRIDE + tid];
    S[tid] = s;
    if (tid == 0) {
        float c = 0.f;
        for (int r = 0; r < NREC2; ++r) c += in[(size_t)r * REC_STRIDE + 512];
        Cs = c;
    }
    __syncthreads();

    if (tid < 128) {
        const float S1 = S[tid];
        const float S2 = S[128 + tid];
        const float A  = S[256 + tid];
        const float Bm = S[384 + tid];
        T[tid] = (2.0f * S1 * Bm - S2 * A) * (1.0f / (float)N_);
    }
    __syncthreads();
    for (int st = 64; st > 0; st >>= 1) {
        if (tid < st) T[tid] += T[tid + st];
        __syncthreads();
    }
    if (tid == 0) out[0] = (-0.5f / (float)N_) * (Cs + T[0]);
}

extern "C" void kernel_launch(void* const* d_in, const int* in_sizes, int n_in,
                              void* d_out, int out_size, void* d_ws, size_t ws_size,
                              hipStream_t stream) {
    (void)in_sizes; (void)n_in; (void)out_size; (void)ws_size;
    const float* x  = (const float*)d_in[0];   // (16,128,64,64) f32
    const float* mu = (const float*)d_in[1];   // (65536,128)   f32
    const float* lv = (const float*)d_in[2];   // (65536,128)   f32
    float* ws = (float*)d_ws;                  // needs ~1.2 MB

    club_k1<<<K1_WGS, 256, 0, stream>>>(x, mu, lv, ws);
    club_k2<<<NREC2, 256, 0, stream>>>(ws, ws + (size_t)K1_WGS * REC_STRIDE);
    club_k3<<<1, 512, 0, stream>>>(ws + (size_t)K1_WGS * REC_STRIDE, (float*)d_out);
}